// SetConv_18794776887560
// MI455X (gfx1250) — compile-verified
//
#include <hip/hip_runtime.h>

// Problem constants (match reference)
#define BB    4
#define NN    8192
#define KK    32
#define CIN   64
#define CF    67      // CIN + 3 edge feats
#define KP    68      // padded K-dim (multiple of 4); col 67 zeroed
#define MID   64
#define COUT  64
#define GRP   8
#define SLOPE 0.1f
#define EPSV  1e-5f

typedef __attribute__((ext_vector_type(2))) float v2f;
typedef __attribute__((ext_vector_type(8))) float v8f;

// ---------------------------------------------------------------------------
// Wave-level 16(rows) x 64(cols) GEMM tile using V_WMMA_F32_16X16X4_F32.
// Xs: [rows][KP] LDS row-major activations. Ws: [64][KP] LDS weights,
// Ws[m][c] = W[out_ch m][in_feat c]  (B fragment element (K=c,N=m)).
// A frag (16x4 f32): lane L<16 -> M=L, VGPR j holds K = k0+j;   lanes>=16
// hold K = k0+2+j (ISA 7.12.2).  B frag mirrors the same K split with
// N = lane%16 striped across lanes.
// ---------------------------------------------------------------------------
static __device__ __forceinline__ void wave_gemm(const float* Xs, const float* Ws,
                                                 int rowBase, int kdim, int lane,
                                                 v8f acc[4]) {
    const int l16  = lane & 15;
    const int koff = (lane >> 4) << 1;            // 0 or 2
    const float* xrow = Xs + (rowBase + l16) * KP;
    for (int k0 = 0; k0 < kdim; k0 += 4) {
        v2f a;
        a.x = xrow[k0 + koff];
        a.y = xrow[k0 + koff + 1];
#pragma unroll
        for (int t = 0; t < 4; ++t) {
            const float* wrow = Ws + (t * 16 + l16) * KP;
            v2f b;
            b.x = wrow[k0 + koff];
            b.y = wrow[k0 + koff + 1];
            acc[t] = __builtin_amdgcn_wmma_f32_16x16x4_f32(
                false, a, false, b, (short)0, acc[t], false, false);
        }
    }
}

// Per-wave (sum, sumsq) reduction into LDS sred[GRP*2].
// Within col-tile t, lanes {0-7,16-23} hold group 2t, lanes {8-15,24-31}
// hold group 2t+1; XOR masks {1,2,4,16} reduce each 16-lane set.
static __device__ __forceinline__ void stats_reduce(v8f acc[4], int lane, float* sred) {
    const int l16 = lane & 15;
#pragma unroll
    for (int t = 0; t < 4; ++t) {
        float s = 0.f, q = 0.f;
#pragma unroll
        for (int v = 0; v < 8; ++v) { float x = acc[t][v]; s += x; q += x * x; }
        s += __shfl_xor(s, 1, 32);  q += __shfl_xor(q, 1, 32);
        s += __shfl_xor(s, 2, 32);  q += __shfl_xor(q, 2, 32);
        s += __shfl_xor(s, 4, 32);  q += __shfl_xor(q, 4, 32);
        s += __shfl_xor(s, 16, 32); q += __shfl_xor(q, 16, 32);
        if ((lane & 7) == 0 && lane < 16) {
            int g = 2 * t + (l16 >> 3);
            atomicAdd(&sred[2 * g + 0], s);
            atomicAdd(&sred[2 * g + 1], q);
        }
    }
}

// ---------------------------------------------------------------------------
// fc1 staging: WG handles 4 points (128 edge rows). Builds in LDS:
//   Xs[r][0:64]  = signal[edges[r]] - signal[center(r)]   (L2-resident gather)
//   Xs[r][64:67] = edge_feats[r],  Xs[r][67] = 0
//   W1s[m][c]    = W1[m][c] (c<67) else 0
// ---------------------------------------------------------------------------
static __device__ __forceinline__ void fc1_stage(const float* __restrict__ signal,
                                                 const int* __restrict__ edges,
                                                 const float* __restrict__ efeat,
                                                 const float* __restrict__ W1,
                                                 float* Xs, float* W1s, float* Cs,
                                                 int p0) {
    const int tid = threadIdx.x;
    {   // center rows for the 4 points (4 x 64 floats)
        int pt = tid >> 6, c = tid & 63;
        Cs[pt * CIN + c] = signal[(long long)(p0 + pt) * CIN + c];
    }
    for (int i = tid; i < MID * KP; i += 256) {   // stage W1 with zero pad col
        int m = i / KP, c = i - m * KP;
        W1s[i] = (c < CF) ? W1[m * CF + c] : 0.f;
    }
    __syncthreads();
    {   // gather: 2 threads per edge row, 8 x float4 each
        int r = tid >> 1;                 // 0..127
        int h = tid & 1;                  // which 32-channel half
        int p = p0 + (r >> 5);
        int k = r & 31;
        int R = p * KK + k;
        int e = edges[R];
        const float4* src = ((const float4*)signal) + (long long)e * 16 + h * 8;
        const float4* ctr = ((const float4*)Cs) + (r >> 5) * 16 + h * 8;
        float4* dst = ((float4*)(Xs + r * KP)) + h * 8;
#pragma unroll
        for (int i = 0; i < 8; ++i) {
            float4 gv = src[i], cv = ctr[i], o;
            o.x = gv.x - cv.x; o.y = gv.y - cv.y;
            o.z = gv.z - cv.z; o.w = gv.w - cv.w;
            dst[i] = o;
        }
        if (h) {
            const float* e3 = efeat + (long long)R * 3;
            float* xr = Xs + r * KP;
            xr[64] = e3[0]; xr[65] = e3[1]; xr[66] = e3[2]; xr[67] = 0.f;
        }
    }
    __syncthreads();
}

// ---------------------------------------------------------------------------
// Pass 1: fc1 + per-(batch,group) sum/sumsq
// ---------------------------------------------------------------------------
__global__ __launch_bounds__(256) void k_fc1_stats(const float* __restrict__ signal,
                                                   const int* __restrict__ edges,
                                                   const float* __restrict__ efeat,
                                                   const float* __restrict__ W1,
                                                   float* __restrict__ stats1) {
    __shared__ __align__(16) float Xs[128 * KP];
    __shared__ __align__(16) float W1s[MID * KP];
    __shared__ __align__(16) float Cs[4 * CIN];
    __shared__ float sred[GRP * 2];
    const int tid = threadIdx.x;
    if (tid < GRP * 2) sred[tid] = 0.f;
    const int p0 = blockIdx.x * 4;                 // 4 points per WG, same batch
    fc1_stage(signal, edges, efeat, W1, Xs, W1s, Cs, p0);

    const int wv = tid >> 5, lane = tid & 31;
    v8f acc[4];
    const v8f z = {0.f, 0.f, 0.f, 0.f, 0.f, 0.f, 0.f, 0.f};
    acc[0] = z; acc[1] = z; acc[2] = z; acc[3] = z;
    wave_gemm(Xs, W1s, wv * 16, KP, lane, acc);

    stats_reduce(acc, lane, sred);
    __syncthreads();
    if (tid < GRP * 2) atomicAdd(&stats1[(p0 >> 13) * GRP * 2 + tid], sred[tid]);
}

// ---------------------------------------------------------------------------
// Pass 2: recompute fc1, apply GN affine + lrelu, maxpool over K=32 -> h1
// ---------------------------------------------------------------------------
__global__ __launch_bounds__(256) void k_fc1_pool(const float* __restrict__ signal,
                                                  const int* __restrict__ edges,
                                                  const float* __restrict__ efeat,
                                                  const float* __restrict__ W1,
                                                  const float* __restrict__ g1,
                                                  const float* __restrict__ b1,
                                                  const float* __restrict__ stats1,
                                                  float* __restrict__ h1) {
    __shared__ __align__(16) float Xs[128 * KP];
    __shared__ __align__(16) float W1s[MID * KP];
    __shared__ __align__(16) float Cs[4 * CIN];
    __shared__ float gam[MID], bet[MID];
    __shared__ float st[GRP * 2];
    __shared__ float tmax[8 * MID];
    const int tid = threadIdx.x;
    const int p0 = blockIdx.x * 4;
    const int b  = p0 >> 13;
    if (tid < MID) { gam[tid] = g1[tid]; bet[tid] = b1[tid]; }
    if (tid < GRP * 2) st[tid] = stats1[b * GRP * 2 + tid];
    fc1_stage(signal, edges, efeat, W1, Xs, W1s, Cs, p0);

    const int wv = tid >> 5, lane = tid & 31, l16 = lane & 15;
    v8f acc[4];
    const v8f z = {0.f, 0.f, 0.f, 0.f, 0.f, 0.f, 0.f, 0.f};
    acc[0] = z; acc[1] = z; acc[2] = z; acc[3] = z;
    wave_gemm(Xs, W1s, wv * 16, KP, lane, acc);

    const float invc = 1.f / (float)(NN * KK * (MID / GRP));   // 1/2097152
#pragma unroll
    for (int t = 0; t < 4; ++t) {
        int c = t * 16 + l16;
        int g = c >> 3;
        float mean = st[2 * g] * invc;
        float var  = st[2 * g + 1] * invc - mean * mean;
        float istd = rsqrtf(var + EPSV);
        float sc = gam[c] * istd;
        float bs = bet[c] - mean * sc;
        float mx = -3.402823466e38f;
#pragma unroll
        for (int v = 0; v < 8; ++v) {
            float x = acc[t][v] * sc + bs;
            x = (x >= 0.f) ? x : SLOPE * x;
            mx = fmaxf(mx, x);
        }
        mx = fmaxf(mx, __shfl_xor(mx, 16, 32));   // combine rows M and M+8
        if (lane < 16) tmax[wv * MID + c] = mx;   // per-16-row-tile max
    }
    __syncthreads();
    {   // combine the 2 row-tiles of each point, write h1
        int q = tid >> 6, c = tid & 63;
        float h = fmaxf(tmax[(2 * q) * MID + c], tmax[(2 * q + 1) * MID + c]);
        h1[(long long)(p0 + q) * MID + c] = h;
    }
}

// ---------------------------------------------------------------------------
// Dense layer (fc2/fc3): 128 rows per WG, K=64. Writes raw output + stats.
// ---------------------------------------------------------------------------
__global__ __launch_bounds__(256) void k_fc_stats(const float* __restrict__ X,
                                                  const float* __restrict__ W,
                                                  float* __restrict__ Yraw,
                                                  float* __restrict__ stats) {
    __shared__ __align__(16) float Xs[128 * KP];
    __shared__ __align__(16) float Ws[MID * KP];
    __shared__ float sred[GRP * 2];
    const int tid = threadIdx.x;
    if (tid < GRP * 2) sred[tid] = 0.f;
    const long long p0 = (long long)blockIdx.x * 128;
    for (int i = tid; i < 128 * 16; i += 256) {        // 128 rows x 16 float4
        int r = i >> 4, c4 = i & 15;
        float4 v = ((const float4*)(X + (p0 + r) * 64))[c4];
        ((float4*)(Xs + r * KP))[c4] = v;
    }
    for (int i = tid; i < MID * KP; i += 256) {
        int m = i / KP, c = i - m * KP;
        Ws[i] = (c < 64) ? W[m * 64 + c] : 0.f;
    }
    __syncthreads();

    const int wv = tid >> 5, lane = tid & 31, l16 = lane & 15, half = lane >> 4;
    v8f acc[4];
    const v8f z = {0.f, 0.f, 0.f, 0.f, 0.f, 0.f, 0.f, 0.f};
    acc[0] = z; acc[1] = z; acc[2] = z; acc[3] = z;
    wave_gemm(Xs, Ws, wv * 16, 64, lane, acc);

#pragma unroll
    for (int t = 0; t < 4; ++t)
#pragma unroll
        for (int v = 0; v < 8; ++v) {
            int row = wv * 16 + v + 8 * half;          // C-layout: M = v + 8*half
            Yraw[(p0 + row) * 64 + t * 16 + l16] = acc[t][v];
        }
    stats_reduce(acc, lane, sred);
    __syncthreads();
    if (tid < GRP * 2) atomicAdd(&stats[(int)(p0 >> 13) * GRP * 2 + tid], sred[tid]);
}

// In-place GroupNorm + lrelu (stats finalized per thread; 2 loads + rsqrt).
__global__ __launch_bounds__(256) void k_gn_apply(float* __restrict__ Y,
                                                  const float* __restrict__ stats,
                                                  const float* __restrict__ gamma,
                                                  const float* __restrict__ beta,
                                                  float inv_count) {
    long long i = (long long)blockIdx.x * 256 + threadIdx.x;
    int c = (int)(i & 63);
    long long p = i >> 6;
    int b = (int)(p >> 13);
    int g = c >> 3;
    float sum = stats[b * GRP * 2 + 2 * g];
    float sq  = stats[b * GRP * 2 + 2 * g + 1];
    float mean = sum * inv_count;
    float var  = sq * inv_count - mean * mean;
    float istd = rsqrtf(var + EPSV);
    float sc = gamma[c] * istd;
    float bs = beta[c] - mean * sc;
    float x = Y[i] * sc + bs;
    Y[i] = (x >= 0.f) ? x : SLOPE * x;
}

__global__ void k_init(float* __restrict__ ws) {
    if (threadIdx.x < 3 * BB * GRP * 2) ws[threadIdx.x] = 0.f;
}

// ---------------------------------------------------------------------------
extern "C" void kernel_launch(void* const* d_in, const int* in_sizes, int n_in,
                              void* d_out, int out_size, void* d_ws, size_t ws_size,
                              hipStream_t stream) {
    (void)in_sizes; (void)n_in; (void)out_size; (void)ws_size;
    const float* signal = (const float*)d_in[0];
    const int*   edges  = (const int*)d_in[1];
    const float* efeat  = (const float*)d_in[2];
    const float* W1 = (const float*)d_in[3];
    const float* g1 = (const float*)d_in[4];
    const float* b1 = (const float*)d_in[5];
    const float* W2 = (const float*)d_in[6];
    const float* g2 = (const float*)d_in[7];
    const float* b2 = (const float*)d_in[8];
    const float* W3 = (const float*)d_in[9];
    const float* g3 = (const float*)d_in[10];
    const float* b3 = (const float*)d_in[11];
    float* out = (float*)d_out;

    float* ws = (float*)d_ws;
    float* stats1 = ws;                  // [4][8][2]
    float* stats2 = ws + 64;
    float* stats3 = ws + 128;
    float* h1   = ws + 256;              // [32768][64]
    float* buf2 = h1 + (long long)32768 * 64;   // [32768][64] (y2 raw -> h2 in place)

    const float invc23 = 1.f / (float)(NN * (COUT / GRP));   // 1/65536

    hipLaunchKernelGGL(k_init,      dim3(1),    dim3(256), 0, stream, ws);
    hipLaunchKernelGGL(k_fc1_stats, dim3(8192), dim3(256), 0, stream,
                       signal, edges, efeat, W1, stats1);
    hipLaunchKernelGGL(k_fc1_pool,  dim3(8192), dim3(256), 0, stream,
                       signal, edges, efeat, W1, g1, b1, stats1, h1);
    hipLaunchKernelGGL(k_fc_stats,  dim3(256),  dim3(256), 0, stream,
                       h1, W2, buf2, stats2);
    hipLaunchKernelGGL(k_gn_apply,  dim3(8192), dim3(256), 0, stream,
                       buf2, stats2, g2, b2, invc23);
    hipLaunchKernelGGL(k_fc_stats,  dim3(256),  dim3(256), 0, stream,
                       buf2, W3, out, stats3);
    hipLaunchKernelGGL(k_gn_apply,  dim3(8192), dim3(256), 0, stream,
                       out, stats3, g3, b3, invc23);
}